// VAE_42021960024333
// MI455X (gfx1250) — compile-verified
//
#include <hip/hip_runtime.h>
#include <cmath>

// ---------------------------------------------------------------------------
// CDNA5 (gfx1250) f32 WMMA: V_WMMA_F32_16X16X4_F32
//   A: 16x4 f32  -> 2 VGPRs/lane (lanes 0-15: M=lane,K=0..1; lanes 16-31: K=2..3)
//   B: 4x16 f32  -> 2 VGPRs/lane (lanes 0-15: N=lane,K=0..1; lanes 16-31: K=2..3)
//   C/D: 16x16 f32 -> 8 VGPRs (VGPR r: lanes 0-15 M=r, lanes 16-31 M=r+8; N=lane&15)
// ---------------------------------------------------------------------------
typedef float v2f __attribute__((ext_vector_type(2)));
typedef float v8f __attribute__((ext_vector_type(8)));

__device__ __forceinline__ v8f wmma4(v2f a, v2f b, v8f c) {
  return __builtin_amdgcn_wmma_f32_16x16x4_f32(
      false, a, false, b, (short)0, c, false, false);
}

__device__ __forceinline__ float eluf(float x) { return x > 0.f ? x : expm1f(x); }
__device__ __forceinline__ float softplusf(float x) {
  return fmaxf(x, 0.f) + log1pf(expf(-fabsf(x)));
}
__device__ __forceinline__ int imin(int a, int b) { return a < b ? a : b; }

__device__ __forceinline__ void storeC(float v, int row, int col, int M, int N,
                                       const float* bias, float* C,
                                       int scR, int scC, int bsR, int bsC,
                                       int act, int accum) {
  if (row < M && col < N) {
    if (bias) v += bias[row * bsR + col * bsC];
    if (act == 1) v = eluf(v);
    const long long ci = (long long)row * scR + (long long)col * scC;
    if (accum) v += C[ci];
    C[ci] = v;
  }
}

// ---------------------------------------------------------------------------
// Generic stride-parameterized WMMA GEMM, 32x32 tile per wave (2x2 WMMA tiles).
//   C[z][m,n] (+)= act( sum_k A[z][m,k]*B[z][k,n] + bias[m*bsR + n*bsC] )
// Branch-free main K loop: row/col indices clamped to last valid element
// (OOB A rows / B cols only feed C entries that are never stored).
// AVEC: A k-stride==1 and row stride even -> b64 fragment loads.
// BVEC: B k-stride==1 and col stride even -> b64 fragment loads.
// ---------------------------------------------------------------------------
template <bool AVEC, bool BVEC>
__global__ void gemm_wmma(const float* __restrict__ A, const float* __restrict__ B,
                          const float* __restrict__ bias, float* __restrict__ C,
                          int M, int N, int K,
                          int saR, int saC, long long batA,
                          int sbR, int sbC, long long batB,
                          int scR, int scC, long long batC,
                          int bsR, int bsC, int act, int accum) {
  const int wave = threadIdx.x >> 5;
  const int lane = threadIdx.x & 31;
  const int tilesN = (N + 31) >> 5;
  const int tilesM = (M + 31) >> 5;
  const int tile = blockIdx.x * 4 + wave;
  if (tile >= tilesM * tilesN) return;           // wave-uniform
  const int tm = (tile / tilesN) << 5;
  const int tn = (tile % tilesN) << 5;
  const long long z = blockIdx.y;
  A += z * batA; B += z * batB; C += z * batC;

  const int l15 = lane & 15;
  const int khalf = (lane >> 4) << 1;            // 0 or 2
  const int m0 = imin(tm + l15, M - 1);
  const int m1 = imin(tm + 16 + l15, M - 1);
  const int n0 = imin(tn + l15, N - 1);
  const int n1 = imin(tn + 16 + l15, N - 1);

  const long long stepA = 4LL * saC;
  const long long stepB = 4LL * sbR;
  const float* pa0 = A + (long long)m0 * saR + (long long)khalf * saC;
  const float* pa0b = pa0 + saC;
  const float* pa1 = A + (long long)m1 * saR + (long long)khalf * saC;
  const float* pa1b = pa1 + saC;
  const float* pb0 = B + (long long)n0 * sbC + (long long)khalf * sbR;
  const float* pb0b = pb0 + sbR;
  const float* pb1 = B + (long long)n1 * sbC + (long long)khalf * sbR;
  const float* pb1b = pb1 + sbR;

  v8f acc00 = {}, acc01 = {}, acc10 = {}, acc11 = {};
  const int kmain = K & ~3;
  int k0 = 0;
  for (; k0 < kmain; k0 += 4) {
    v2f a0, a1, b0, b1;
    if (AVEC) {
      a0 = *(const v2f*)pa0;
      a1 = *(const v2f*)pa1;
    } else {
      a0.x = *pa0;  a0.y = *pa0b;
      a1.x = *pa1;  a1.y = *pa1b;
      pa0b += stepA; pa1b += stepA;
    }
    if (BVEC) {
      b0 = *(const v2f*)pb0;
      b1 = *(const v2f*)pb1;
    } else {
      b0.x = *pb0;  b0.y = *pb0b;
      b1.x = *pb1;  b1.y = *pb1b;
      pb0b += stepB; pb1b += stepB;
    }
    pa0 += stepA; pa1 += stepA; pb0 += stepB; pb1 += stepB;
    acc00 = wmma4(a0, b0, acc00);
    acc01 = wmma4(a0, b1, acc01);
    acc10 = wmma4(a1, b0, acc10);
    acc11 = wmma4(a1, b1, acc11);
  }
  if (k0 < K) {                                  // K-tail: clamp addr, zero value
    const int ka = k0 + khalf;
    const int kc0 = imin(ka, K - 1);
    const int kc1 = imin(ka + 1, K - 1);
    const bool v0 = ka < K, v1 = (ka + 1) < K;
    float t;
    v2f a0, a1, b0, b1;
    t = A[(long long)m0 * saR + (long long)kc0 * saC]; a0.x = v0 ? t : 0.f;
    t = A[(long long)m0 * saR + (long long)kc1 * saC]; a0.y = v1 ? t : 0.f;
    t = A[(long long)m1 * saR + (long long)kc0 * saC]; a1.x = v0 ? t : 0.f;
    t = A[(long long)m1 * saR + (long long)kc1 * saC]; a1.y = v1 ? t : 0.f;
    t = B[(long long)kc0 * sbR + (long long)n0 * sbC]; b0.x = v0 ? t : 0.f;
    t = B[(long long)kc1 * sbR + (long long)n0 * sbC]; b0.y = v1 ? t : 0.f;
    t = B[(long long)kc0 * sbR + (long long)n1 * sbC]; b1.x = v0 ? t : 0.f;
    t = B[(long long)kc1 * sbR + (long long)n1 * sbC]; b1.y = v1 ? t : 0.f;
    acc00 = wmma4(a0, b0, acc00);
    acc01 = wmma4(a0, b1, acc01);
    acc10 = wmma4(a1, b0, acc10);
    acc11 = wmma4(a1, b1, acc11);
  }

  const int cA = tn + l15, cB = cA + 16;
  const int rb = (lane >> 4) << 3;
#pragma unroll
  for (int r = 0; r < 8; ++r) {
    const int r0 = tm + rb + r, r1 = r0 + 16;
    storeC(acc00[r], r0, cA, M, N, bias, C, scR, scC, bsR, bsC, act, accum);
    storeC(acc01[r], r0, cB, M, N, bias, C, scR, scC, bsR, bsC, act, accum);
    storeC(acc10[r], r1, cA, M, N, bias, C, scR, scC, bsR, bsC, act, accum);
    storeC(acc11[r], r1, cB, M, N, bias, C, scR, scC, bsR, bsC, act, accum);
  }
}

// ---------------------------------------------------------------------------
// Scan layer 1: h1 = ELU( [z1 | y | sin t | cos t] @ W0(138x256) + b0 )
// Three branch-free K phases: [0,8) from zcat (b64 loads), [8,136) from ycur
// (b64 loads), [136,140) = {sin, cos, 0, 0}. M=N=256, 32x32 tile -> 64 tiles.
// ---------------------------------------------------------------------------
__global__ void scan_l1(const float* __restrict__ zcat, const float* __restrict__ ycur,
                        const float* __restrict__ W0, const float* __restrict__ b0c,
                        float* __restrict__ h1, float st, float ct) {
  const int wave = threadIdx.x >> 5;
  const int lane = threadIdx.x & 31;
  const int tile = blockIdx.x * 4 + wave;        // 64 tiles (8x8)
  const int tm = (tile >> 3) << 5;
  const int tn = (tile & 7) << 5;
  const int l15 = lane & 15;
  const int khalf = (lane >> 4) << 1;
  const int m0 = tm + l15, m1 = m0 + 16;
  const int n0 = tn + l15, n1 = n0 + 16;

  v8f acc00 = {}, acc01 = {}, acc10 = {}, acc11 = {};

  // phase 1: k in [0,8) -> zcat[:, 0:8] (row stride 16; (m*16+ka) even -> b64)
#pragma unroll
  for (int k0 = 0; k0 < 8; k0 += 4) {
    const int ka = k0 + khalf;
    v2f a0, a1, b0, b1;
    a0 = *(const v2f*)(zcat + m0 * 16 + ka);
    a1 = *(const v2f*)(zcat + m1 * 16 + ka);
    b0.x = W0[ka * 256 + n0];      b0.y = W0[(ka + 1) * 256 + n0];
    b1.x = W0[ka * 256 + n1];      b1.y = W0[(ka + 1) * 256 + n1];
    acc00 = wmma4(a0, b0, acc00);
    acc01 = wmma4(a0, b1, acc01);
    acc10 = wmma4(a1, b0, acc10);
    acc11 = wmma4(a1, b1, acc11);
  }

  // phase 2: k in [8,136) -> ycur (row stride 128); W0 rows 8..135
  {
    const float* pa0 = ycur + m0 * 128 + khalf;
    const float* pa1 = ycur + m1 * 128 + khalf;
    const float* pw0 = W0 + (8 + khalf) * 256 + n0;
    const float* pw1 = W0 + (8 + khalf) * 256 + n1;
    for (int kk = 0; kk < 128; kk += 4) {
      v2f a0, a1, b0, b1;
      a0 = *(const v2f*)pa0;
      a1 = *(const v2f*)pa1;
      b0.x = pw0[0]; b0.y = pw0[256];
      b1.x = pw1[0]; b1.y = pw1[256];
      pa0 += 4; pa1 += 4; pw0 += 4 * 256; pw1 += 4 * 256;
      acc00 = wmma4(a0, b0, acc00);
      acc01 = wmma4(a0, b1, acc01);
      acc10 = wmma4(a1, b0, acc10);
      acc11 = wmma4(a1, b1, acc11);
    }
  }

  // phase 3: k = 136 (sin), 137 (cos), 138/139 zero-pad
  {
    const bool lo = (khalf == 0);
    v2f a0, a1, b0, b1;
    a0.x = lo ? st : 0.f;  a0.y = lo ? ct : 0.f;
    a1 = a0;
    const int kc = lo ? 136 : 137;               // clamped row (value zeroed below)
    float t0 = W0[kc * 256 + n0], t0b = W0[137 * 256 + n0];
    float t1 = W0[kc * 256 + n1], t1b = W0[137 * 256 + n1];
    b0.x = lo ? t0 : 0.f;  b0.y = lo ? t0b : 0.f;
    b1.x = lo ? t1 : 0.f;  b1.y = lo ? t1b : 0.f;
    acc00 = wmma4(a0, b0, acc00);
    acc01 = wmma4(a0, b1, acc01);
    acc10 = wmma4(a1, b0, acc10);
    acc11 = wmma4(a1, b1, acc11);
  }

  const int cA = tn + l15, cB = cA + 16;
  const int rb = (lane >> 4) << 3;
  const float bA = b0c[cA], bB = b0c[cB];
#pragma unroll
  for (int r = 0; r < 8; ++r) {
    const int r0 = tm + rb + r, r1 = r0 + 16;
    h1[r0 * 256 + cA] = eluf(acc00[r] + bA);
    h1[r0 * 256 + cB] = eluf(acc01[r] + bB);
    h1[r1 * 256 + cA] = eluf(acc10[r] + bA);
    h1[r1 * 256 + cB] = eluf(acc11[r] + bB);
  }
}

// ---------------------------------------------------------------------------
// Scan layer 3 + Euler update:  out = h2 @ W2(256x128) + b2
//   ynext = y + DT_INTG*(dcoeff * (y[c-1]-2y[c]+y[c+1]) / dx^2 + out)
// M=256, N=128, 32x32 tile per wave -> 32 tiles.
// ---------------------------------------------------------------------------
__global__ void scan_l3(const float* __restrict__ h2, const float* __restrict__ W2,
                        const float* __restrict__ b2, const float* __restrict__ ycur,
                        const float* __restrict__ dcf, float* __restrict__ ynext,
                        float* __restrict__ yhist) {
  const int wave = threadIdx.x >> 5;
  const int lane = threadIdx.x & 31;
  const int tile = blockIdx.x * 4 + wave;        // 32 tiles (8 x 4)
  const int tm = (tile >> 2) << 5;
  const int tn = (tile & 3) << 5;
  const int l15 = lane & 15;
  const int khalf = (lane >> 4) << 1;
  const int m0 = tm + l15, m1 = m0 + 16;
  const int n0 = tn + l15, n1 = n0 + 16;

  const float* pa0 = h2 + m0 * 256 + khalf;
  const float* pa1 = h2 + m1 * 256 + khalf;
  const float* pw0 = W2 + khalf * 128 + n0;
  const float* pw1 = W2 + khalf * 128 + n1;

  v8f acc00 = {}, acc01 = {}, acc10 = {}, acc11 = {};
  for (int k0 = 0; k0 < 256; k0 += 4) {
    v2f a0, a1, b0, b1;
    a0 = *(const v2f*)pa0;
    a1 = *(const v2f*)pa1;
    b0.x = pw0[0]; b0.y = pw0[128];
    b1.x = pw1[0]; b1.y = pw1[128];
    pa0 += 4; pa1 += 4; pw0 += 4 * 128; pw1 += 4 * 128;
    acc00 = wmma4(a0, b0, acc00);
    acc01 = wmma4(a0, b1, acc01);
    acc10 = wmma4(a1, b0, acc10);
    acc11 = wmma4(a1, b1, acc11);
  }

  const int cA = tn + l15, cB = cA + 16;
  const int rb = (lane >> 4) << 3;
  const float b2A = b2[cA], b2B = b2[cB];
#pragma unroll
  for (int r = 0; r < 8; ++r) {
    const int r0 = tm + rb + r, r1 = r0 + 16;
    const float d0 = dcf[r0], d1 = dcf[r1];
#pragma unroll
    for (int q = 0; q < 2; ++q) {
      const int nb = q ? r1 : r0;
      const float dc = q ? d1 : d0;
      {
        const float o  = (q ? acc10[r] : acc00[r]) + b2A;
        const float y0 = ycur[nb * 128 + cA];
        const float yl = (cA > 0)   ? ycur[nb * 128 + cA - 1] : 0.f;
        const float yr = (cA < 127) ? ycur[nb * 128 + cA + 1] : 0.f;
        const float yv = y0 + 0.005f * (dc * (yl - 2.f * y0 + yr) * 100.f + o);
        ynext[nb * 128 + cA] = yv;
        if (yhist) yhist[(long long)nb * 32000 + cA] = yv;
      }
      {
        const float o  = (q ? acc11[r] : acc01[r]) + b2B;
        const float y0 = ycur[nb * 128 + cB];
        const float yl = ycur[nb * 128 + cB - 1];
        const float yr = (cB < 127) ? ycur[nb * 128 + cB + 1] : 0.f;
        const float yv = y0 + 0.005f * (dc * (yl - 2.f * y0 + yr) * 100.f + o);
        ynext[nb * 128 + cB] = yv;
        if (yhist) yhist[(long long)nb * 32000 + cB] = yv;
      }
    }
  }
}

// ---------------------------------------------------------------------------
// Scalar helpers for the tiny heads (N < 16)
// ---------------------------------------------------------------------------
__global__ void linear_small(const float* __restrict__ A, const float* __restrict__ W,
                             const float* __restrict__ b, float* __restrict__ out,
                             int M, int K, int Nout, int lda, int ldc, int act) {
  const int i = blockIdx.x * blockDim.x + threadIdx.x;
  if (i >= M * Nout) return;
  const int m = i / Nout, j = i % Nout;
  float s = b[j];
  for (int k = 0; k < K; ++k) s += A[m * lda + k] * W[k * Nout + j];
  if (act == 2) s = softplusf(s);
  out[m * ldc + j] = s;
}

// cat2 padded to row stride 20 (cols 17..19 zero) so the GEMM A-side stays
// 8-byte aligned for b64 fragment loads.
__global__ void pack_cat2(const float* __restrict__ dcf, const float* __restrict__ zcat,
                          float* __restrict__ cat2) {
  const int i = blockIdx.x * blockDim.x + threadIdx.x;
  if (i >= 256 * 20) return;
  const int m = i / 20, j = i % 20;
  float v = 0.f;
  if (j == 0) v = dcf[m];
  else if (j <= 16) v = zcat[m * 16 + (j - 1)];
  cat2[i] = v;
}

__global__ void y_init(const float* __restrict__ y, float* __restrict__ ya,
                       float* __restrict__ yseq) {
  const int i = blockIdx.x * blockDim.x + threadIdx.x;
  if (i >= 256 * 128) return;
  const int n = i >> 7, c = i & 127;
  const float v = y[(long long)n * 32000 + c * 250];
  ya[i] = v;
  yseq[(long long)n * 32000 + c] = v;    // y_seq[n, 0, c]
}

// ---------------------------------------------------------------------------
static inline void launch_gemm(hipStream_t s, const float* A, const float* B,
                               const float* bias, float* C, int M, int N, int K,
                               int saR, int saC, long long batA,
                               int sbR, int sbC, long long batB,
                               int scR, int scC, long long batC,
                               int bsR, int bsC, int act, int accum, int batch) {
  const int tiles = ((M + 31) / 32) * ((N + 31) / 32);
  dim3 grid((tiles + 3) / 4, batch, 1);
  const bool av = (saC == 1) && ((saR & 1) == 0);
  const bool bv = (sbR == 1) && ((sbC & 1) == 0);
  if (av && bv)
    gemm_wmma<true, true><<<grid, 128, 0, s>>>(A, B, bias, C, M, N, K, saR, saC, batA,
        sbR, sbC, batB, scR, scC, batC, bsR, bsC, act, accum);
  else if (av)
    gemm_wmma<true, false><<<grid, 128, 0, s>>>(A, B, bias, C, M, N, K, saR, saC, batA,
        sbR, sbC, batB, scR, scC, batC, bsR, bsC, act, accum);
  else if (bv)
    gemm_wmma<false, true><<<grid, 128, 0, s>>>(A, B, bias, C, M, N, K, saR, saC, batA,
        sbR, sbC, batB, scR, scC, batC, bsR, bsC, act, accum);
  else
    gemm_wmma<false, false><<<grid, 128, 0, s>>>(A, B, bias, C, M, N, K, saR, saC, batA,
        sbR, sbC, batB, scR, scC, batC, bsR, bsC, act, accum);
}

extern "C" void kernel_launch(void* const* d_in, const int* in_sizes, int n_in,
                              void* d_out, int out_size, void* d_ws, size_t ws_size,
                              hipStream_t stream) {
  const float* x     = (const float*)d_in[0];
  const float* Wf1_0 = (const float*)d_in[1];  const float* bf1_0 = (const float*)d_in[2];
  const float* Wf1_1 = (const float*)d_in[3];  const float* bf1_1 = (const float*)d_in[4];
  const float* Wz1_0 = (const float*)d_in[5];  const float* bz1_0 = (const float*)d_in[6];
  const float* Wz1_1 = (const float*)d_in[7];  const float* bz1_1 = (const float*)d_in[8];
  const float* Wf2_0 = (const float*)d_in[9];  const float* bf2_0 = (const float*)d_in[10];
  const float* Wf2_1 = (const float*)d_in[11]; const float* bf2_1 = (const float*)d_in[12];
  const float* Wz2_0 = (const float*)d_in[13]; const float* bz2_0 = (const float*)d_in[14];
  const float* Wz2_1 = (const float*)d_in[15]; const float* bz2_1 = (const float*)d_in[16];
  const float* Wmap  = (const float*)d_in[17]; const float* bmap  = (const float*)d_in[18];
  const float* Wur0  = (const float*)d_in[19]; const float* bur0  = (const float*)d_in[20];
  const float* Wur1  = (const float*)d_in[21]; const float* bur1  = (const float*)d_in[22];
  const float* Wfp0  = (const float*)d_in[23]; const float* bfp0  = (const float*)d_in[24];
  const float* Wfp1  = (const float*)d_in[25]; const float* bfp1  = (const float*)d_in[26];
  const float* Wdc0  = (const float*)d_in[27]; const float* bdc0  = (const float*)d_in[28];
  const float* Wdc1  = (const float*)d_in[29]; const float* bdc1  = (const float*)d_in[30];
  const float* Wda0  = (const float*)d_in[31]; const float* bda0  = (const float*)d_in[32];
  const float* Wda1  = (const float*)d_in[33]; const float* bda1  = (const float*)d_in[34];
  const float* Wda2  = (const float*)d_in[35]; const float* bda2  = (const float*)d_in[36];
  const float* Wdmap = (const float*)d_in[37]; const float* bdmap = (const float*)d_in[38];
  const float* Wdr0  = (const float*)d_in[39]; const float* bdr0  = (const float*)d_in[40];
  const float* Wdr1  = (const float*)d_in[41]; const float* bdr1  = (const float*)d_in[42];

  char* ws = (char*)d_ws;
  float* y    = (float*)(ws + 0);          // (256, 128, 250)  32,768,000 B
  float* yseq = (float*)(ws + 32768000);   // (256, 250, 128)  32,768,000 B
  float* f1   = (float*)(ws + 65536000);   // 256x256 temp
  float* f2   = (float*)(ws + 65798144);   // 256x256 temp
  float* h1   = (float*)(ws + 66060288);   // scan hidden 1
  float* h2   = (float*)(ws + 66322432);   // scan hidden 2
  float* zcat = (float*)(ws + 66584576);   // 256x16  [z1|z2]
  float* cat2 = (float*)(ws + 66600960);   // 256x20  [dcoeff|z1|z2|pad]
  float* dcf  = (float*)(ws + 66621440);   // 256 dcoeff
  float* ya   = (float*)(ws + 66622464);   // 256x128 ODE state A
  float* yb   = (float*)(ws + 66753536);   // 256x128 ODE state B
  float* out  = (float*)d_out;

  // ---- aux encoders -> z1, z2 (packed into zcat) ----
  launch_gemm(stream, x,  Wf1_0, bf1_0, f1, 256, 256, 32000, 32000,1,0, 256,1,0, 256,1,0, 0,1, 1,0, 1);
  launch_gemm(stream, f1, Wf1_1, bf1_1, f2, 256, 256, 256,   256,1,0,   256,1,0, 256,1,0, 0,1, 1,0, 1);
  launch_gemm(stream, f2, Wz1_0, bz1_0, f1, 256, 128, 256,   256,1,0,   128,1,0, 128,1,0, 0,1, 1,0, 1);
  linear_small<<<(256*8+127)/128, 128, 0, stream>>>(f1, Wz1_1, bz1_1, zcat,     256, 128, 8, 128, 16, 0);
  launch_gemm(stream, x,  Wf2_0, bf2_0, f1, 256, 256, 32000, 32000,1,0, 256,1,0, 256,1,0, 0,1, 1,0, 1);
  launch_gemm(stream, f1, Wf2_1, bf2_1, f2, 256, 256, 256,   256,1,0,   256,1,0, 256,1,0, 0,1, 1,0, 1);
  launch_gemm(stream, f2, Wz2_0, bz2_0, f1, 256, 128, 256,   256,1,0,   128,1,0, 128,1,0, 0,1, 1,0, 1);
  linear_small<<<(256*8+127)/128, 128, 0, stream>>>(f1, Wz2_1, bz2_1, zcat + 8, 256, 128, 8, 128, 16, 0);

  // ---- unmixer: y[n,c,t] = Wmap^T @ x[n] + bias(row), then += res MLP ----
  launch_gemm(stream, Wmap, x, bmap, y, 128, 250, 128, 1,128,0, 250,1,32000, 250,1,32000, 1,0, 0,0, 256);
  launch_gemm(stream, zcat, Wur0, bur0, f1, 256, 256, 16,   16,1,0,  256,1,0,   256,1,0,   0,1, 1,0, 1);
  launch_gemm(stream, f1,  Wur1, bur1, y,  256, 32000, 256, 256,1,0, 32000,1,0, 32000,1,0, 0,1, 0,1, 1);

  // ---- feat_phy -> dcoeff = softplus(...) ----
  launch_gemm(stream, y,  Wfp0, bfp0, f1, 256, 256, 32000, 32000,1,0, 256,1,0, 256,1,0, 0,1, 1,0, 1);
  launch_gemm(stream, f1, Wfp1, bfp1, f2, 256, 256, 256,   256,1,0,   256,1,0, 256,1,0, 0,1, 1,0, 1);
  launch_gemm(stream, f2, Wdc0, bdc0, f1, 256, 128, 256,   256,1,0,   128,1,0, 128,1,0, 0,1, 1,0, 1);
  linear_small<<<(256+127)/128, 128, 0, stream>>>(f1, Wdc1, bdc1, dcf, 256, 128, 1, 128, 1, 2);
  pack_cat2<<<(256*20+127)/128, 128, 0, stream>>>(dcf, zcat, cat2);

  // ---- ODE scan: 498 Euler steps, record every other state ----
  y_init<<<(256*128+127)/128, 128, 0, stream>>>(y, ya, yseq);
  float* yc = ya; float* yn = yb;
  for (int s = 0; s < 498; ++s) {
    const float tc = (float)(0.005 * (double)s);
    const float st = sinf(tc), ct = cosf(tc);
    scan_l1<<<16, 128, 0, stream>>>(zcat, yc, Wda0, bda0, h1, st, ct);
    launch_gemm(stream, h1, Wda1, bda1, h2, 256, 256, 256, 256,1,0, 256,1,0, 256,1,0, 0,1, 1,0, 1);
    float* hist = (s & 1) ? (yseq + (long long)((s + 1) / 2) * 128) : nullptr;
    scan_l3<<<8, 128, 0, stream>>>(h2, Wda2, bda2, yc, dcf, yn, hist);
    float* t = yc; yc = yn; yn = t;
  }

  // ---- decoder: out = dec_aux2_res(cat2) + Wdmap^T @ y_seq[n]^T ----
  launch_gemm(stream, cat2, Wdr0, bdr0, f1, 256, 256, 17,    20,1,0,  256,1,0,   256,1,0,   0,1, 1,0, 1);
  launch_gemm(stream, f1,  Wdr1, bdr1, out, 256, 32000, 256, 256,1,0, 32000,1,0, 32000,1,0, 0,1, 0,0, 1);
  launch_gemm(stream, Wdmap, yseq, bdmap, out, 128, 250, 128, 1,128,0, 1,128,32000, 250,1,32000, 1,0, 0,1, 256);
}